// graph_inverseSolveNet_43284680409676
// MI455X (gfx1250) — compile-verified
//
#include <hip/hip_runtime.h>
#include <math.h>

#define C 32

typedef float v2f __attribute__((ext_vector_type(2)));
typedef float v8f __attribute__((ext_vector_type(8)));

static __device__ __forceinline__ v8f wmma_f32_4(v2f a, v2f b, v8f c) {
  // D(16x16,f32) = A(16x4,f32) x B(4x16,f32) + C
  return __builtin_amdgcn_wmma_f32_16x16x4_f32(false, a, false, b, (short)0, c, false, false);
}

static __device__ __forceinline__ void wait_lds() {
  asm volatile("s_wait_dscnt 0x0" ::: "memory");
}

// Branch-free tanh: (e^{2x}-1)/(e^{2x}+1); clamp at +-9 (tanh(9)==1.0f in f32).
// Lowers to v_exp_f32 + v_rcp_f32 transcendentals, no EXEC divergence.
static __device__ __forceinline__ float tanh_fast(float x) {
  float xc = fminf(fmaxf(x, -9.f), 9.f);
  float e  = __expf(2.f * xc);
  return (e - 1.f) * __builtin_amdgcn_rcpf(e + 1.f);
}

// ---------------- small utility kernels ----------------

__global__ void k_zero(float4* p, int n4) {
  int t = blockIdx.x * blockDim.x + threadIdx.x;
  if (t < n4) p[t] = make_float4(0.f, 0.f, 0.f, 0.f);
}

// accum[src[e]] += xE[e]  (segment_sum of the initial data_proj)
__global__ void k_scatter_xe(const float4* __restrict__ xE4, const int* __restrict__ src,
                             float* __restrict__ accum, int E) {
  int t = blockIdx.x * blockDim.x + threadIdx.x;
  if (t >= E * 8) return;
  int e = t >> 3, q = t & 7;
  float4 v = xE4[t];
  float* a = accum + src[e] * C + q * 4;
  atomicAdd(a + 0, v.x); atomicAdd(a + 1, v.y);
  atomicAdd(a + 2, v.z); atomicAdd(a + 3, v.w);
}

// R = D - accum ; accum = 0 (ready for next segment_sum) ; optional copy of R to output
__global__ void k_node(const float4* __restrict__ D4, float4* __restrict__ acc4,
                       float4* __restrict__ R4, float4* __restrict__ out4, int n4) {
  int t = blockIdx.x * blockDim.x + threadIdx.x;
  if (t >= n4) return;
  float4 d = D4[t], a = acc4[t];
  float4 r = make_float4(d.x - a.x, d.y - a.y, d.z - a.z, d.w - a.w);
  R4[t] = r;
  acc4[t] = make_float4(0.f, 0.f, 0.f, 0.f);
  if (out4) out4[t] = r;
}

// Z = Zref + ew * (R[src] - R[dst])
__global__ void k_edge_update(const float4* __restrict__ Zref4, const float* __restrict__ ew,
                              const int* __restrict__ src, const int* __restrict__ dst,
                              const float4* __restrict__ R4, float4* __restrict__ Z4, int E) {
  int t = blockIdx.x * blockDim.x + threadIdx.x;
  if (t >= E * 8) return;
  int e = t >> 3, q = t & 7;
  int s = src[e], d = dst[e];
  float w = ew[e];
  float4 rs = R4[s * 8 + q];
  float4 rd = R4[d * 8 + q];
  float4 zr = Zref4[t];
  Z4[t] = make_float4(zr.x + w * (rs.x - rd.x),
                      zr.y + w * (rs.y - rd.y),
                      zr.z + w * (rs.z - rd.z),
                      zr.w + w * (rs.w - rd.w));
}

// ---------------- WMMA edge MLP ----------------
// Per wave: 16 edges. Zref = tanh(G@Wg)@WgT + tanh(S@Ws)@WsT,
// with G = ew*(Z[src]-Z[dst]), S = ew*Z[dst]; also fused atomic segment_sum into accum[src].
__launch_bounds__(128)
__global__ void k_edge_mlp(const float* __restrict__ Z, const int* __restrict__ src,
                           const int* __restrict__ dst, const float* __restrict__ ew,
                           const float* __restrict__ Wg, const float* __restrict__ Ws,
                           float* __restrict__ Zref, float* __restrict__ accum, int E) {
  // B fragments pre-packed as K-pairs: sW[m][p][n] = {W[2p][n], W[2p+1][n]}
  // m: 0=Wg, 1=WgT, 2=Ws, 3=WsT
  __shared__ v2f   sW[4][16][32];
  __shared__ float sIN[4][2][16][36];   // [wave][G/S][edge-row][col], padded rows
  __shared__ float sH[4][16][36];       // tanh intermediate, per wave

  for (int idx = threadIdx.x; idx < 4 * 16 * 32; idx += blockDim.x) {
    int m = idx >> 9, rem = idx & 511, p = rem >> 5, n = rem & 31;
    const float* W = (m < 2) ? Wg : Ws;
    v2f v;
    if (m & 1) { v.x = W[n * C + 2 * p];     v.y = W[n * C + 2 * p + 1]; }   // transposed
    else       { v.x = W[(2 * p) * C + n];   v.y = W[(2 * p + 1) * C + n]; }
    sW[m][p][n] = v;
  }
  __syncthreads();

  int wave = threadIdx.x >> 5, lane = threadIdx.x & 31;
  int h = lane >> 4, l = lane & 15;
  int base = (blockIdx.x * 4 + wave) * 16;
  if (base >= E) return;            // wave-uniform: EXEC stays all-ones inside

  int   sidx = src[base + l];       // lanes 16-31 mirror lanes 0-15 (l = lane&15)
  int   didx = dst[base + l];
  float wgt  = ew[base + l];

  // Build G and S tiles (coalesced 128B row reads; node region is L2-resident)
  for (int e = 0; e < 16; ++e) {
    int   se = __shfl(sidx, e, 32);
    int   de = __shfl(didx, e, 32);
    float we = __shfl(wgt,  e, 32);
    float zs = Z[se * C + lane];
    float zd = Z[de * C + lane];
    sIN[wave][0][e][lane] = we * (zs - zd);
    sIN[wave][1][e][lane] = we * zd;
  }
  wait_lds();

  const v8f vz = {0.f, 0.f, 0.f, 0.f, 0.f, 0.f, 0.f, 0.f};
  v8f zr0 = vz, zr1 = vz;           // Zref tile accumulators (both paths summed)

  for (int path = 0; path < 2; ++path) {
    const float (*IN)[36] = sIN[wave][path];
    const v2f (*B1)[32] = (path == 0) ? sW[0] : sW[2];
    const v2f (*B2)[32] = (path == 0) ? sW[1] : sW[3];

    v8f c0 = vz, c1 = vz;
#pragma unroll
    for (int kb = 0; kb < 8; ++kb) {
      v2f a = *(const v2f*)&IN[l][kb * 4 + 2 * h];
      c0 = wmma_f32_4(a, B1[kb * 2 + h][l],      c0);
      c1 = wmma_f32_4(a, B1[kb * 2 + h][16 + l], c1);
    }
#pragma unroll
    for (int r = 0; r < 8; ++r) { c0[r] = tanh_fast(c0[r]); c1[r] = tanh_fast(c1[r]); }

    wait_lds();                     // prior-path sH reads complete before overwrite
#pragma unroll
    for (int r = 0; r < 8; ++r) {   // C-layout (M=r+8h, N=lane) -> row-major H
      sH[wave][r + 8 * h][l]      = c0[r];
      sH[wave][r + 8 * h][16 + l] = c1[r];
    }
    wait_lds();
#pragma unroll
    for (int kb = 0; kb < 8; ++kb) {
      v2f a = *(const v2f*)&sH[wave][l][kb * 4 + 2 * h];
      zr0 = wmma_f32_4(a, B2[kb * 2 + h][l],      zr0);
      zr1 = wmma_f32_4(a, B2[kb * 2 + h][16 + l], zr1);
    }
  }

  // Write Zref and fuse atomic segment_sum into accum[src]
#pragma unroll
  for (int r = 0; r < 8; ++r) {
    int row = base + r + 8 * h;
    Zref[row * C + l]      = zr0[r];
    Zref[row * C + 16 + l] = zr1[r];
    int srow = __shfl(sidx, r + 8 * h, 32);
    atomicAdd(&accum[srow * C + l],      zr0[r]);
    atomicAdd(&accum[srow * C + 16 + l], zr1[r]);
  }
}

// ---------------- final output GEMMs ----------------
__launch_bounds__(256)
__global__ void k_out_gemm(const float* __restrict__ Z, const float* __restrict__ Zref,
                           const float* __restrict__ We, float* __restrict__ X,
                           float* __restrict__ Xref, int E) {
  __shared__ v2f   sWe[16][32];       // W_emb packed K-pairs
  __shared__ float sB[8][16][36];

  for (int idx = threadIdx.x; idx < 512; idx += blockDim.x) {
    int p = idx >> 5, n = idx & 31;
    v2f v; v.x = We[(2 * p) * C + n]; v.y = We[(2 * p + 1) * C + n];
    sWe[p][n] = v;
  }
  __syncthreads();

  int wave = threadIdx.x >> 5, lane = threadIdx.x & 31;
  int h = lane >> 4, l = lane & 15;
  int base = (blockIdx.x * 8 + wave) * 16;
  if (base >= E) return;

  const v8f vz = {0.f, 0.f, 0.f, 0.f, 0.f, 0.f, 0.f, 0.f};
  for (int m = 0; m < 2; ++m) {
    const float* M = m ? Zref : Z;
    float*       O = m ? Xref : X;

    wait_lds();                       // prior reads done before overwriting sB
    for (int e = 0; e < 16; ++e)
      sB[wave][e][lane] = M[(base + e) * C + lane];
    wait_lds();

    v8f c0 = vz, c1 = vz;
#pragma unroll
    for (int kb = 0; kb < 8; ++kb) {
      v2f a = *(const v2f*)&sB[wave][l][kb * 4 + 2 * h];
      c0 = wmma_f32_4(a, sWe[kb * 2 + h][l],      c0);
      c1 = wmma_f32_4(a, sWe[kb * 2 + h][16 + l], c1);
    }
#pragma unroll
    for (int r = 0; r < 8; ++r) {
      O[(base + r + 8 * h) * C + l]      = c0[r];
      O[(base + r + 8 * h) * C + 16 + l] = c1[r];
    }
  }
}

// ---------------- launcher ----------------
extern "C" void kernel_launch(void* const* d_in, const int* in_sizes, int n_in,
                              void* d_out, int out_size, void* d_ws, size_t ws_size,
                              hipStream_t stream) {
  const float* D  = (const float*)d_in[0];
  const float* xE = (const float*)d_in[1];
  const float* ew = (const float*)d_in[2];
  const float* Wg = (const float*)d_in[3];
  const float* Ws = (const float*)d_in[4];
  const float* We = (const float*)d_in[5];
  const int*   ei = (const int*)d_in[6];

  int E = in_sizes[2];          // edge count
  int N = in_sizes[0] / C;      // node count
  const int* src = ei;
  const int* dst = ei + E;

  float* ws    = (float*)d_ws;
  float* Z     = ws;
  float* Zref  = Z     + (size_t)E * C;
  float* accum = Zref  + (size_t)E * C;
  float* R     = accum + (size_t)N * C;

  float* outX    = (float*)d_out;
  float* outXref = outX    + (size_t)E * C;
  float* outR    = outXref + (size_t)E * C;

  int nc4 = N * C / 4;
  int e8  = E * 8;
  int gz   = (nc4 + 255) / 256;
  int ge   = (e8 + 255) / 256;
  int gmlp = (E + 63) / 64;     // 4 waves x 16 edges per block
  int gout = (E + 127) / 128;   // 8 waves x 16 edges per block

  // data_proj(xE): R = D - segsum(xE, src); Z = xE + ew*(R[src]-R[dst])
  hipLaunchKernelGGL(k_zero, dim3(gz), dim3(256), 0, stream, (float4*)accum, nc4);
  hipLaunchKernelGGL(k_scatter_xe, dim3(ge), dim3(256), 0, stream,
                     (const float4*)xE, src, accum, E);
  hipLaunchKernelGGL(k_node, dim3(gz), dim3(256), 0, stream,
                     (const float4*)D, (float4*)accum, (float4*)R, (float4*)nullptr, nc4);
  hipLaunchKernelGGL(k_edge_update, dim3(ge), dim3(256), 0, stream,
                     (const float4*)xE, ew, src, dst, (const float4*)R, (float4*)Z, E);

  for (int it = 0; it < 3; ++it) {
    hipLaunchKernelGGL(k_edge_mlp, dim3(gmlp), dim3(128), 0, stream,
                       Z, src, dst, ew, Wg, Ws, Zref, accum, E);
    hipLaunchKernelGGL(k_node, dim3(gz), dim3(256), 0, stream,
                       (const float4*)D, (float4*)accum, (float4*)R,
                       (float4*)(it == 2 ? outR : nullptr), nc4);
    hipLaunchKernelGGL(k_edge_update, dim3(ge), dim3(256), 0, stream,
                       (const float4*)Zref, ew, src, dst, (const float4*)R, (float4*)Z, E);
  }

  hipLaunchKernelGGL(k_out_gemm, dim3(gout), dim3(256), 0, stream,
                     Z, Zref, We, outX, outXref, E);
}